// SingleTARNet_23313082482709
// MI455X (gfx1250) — compile-verified
//
#include <hip/hip_runtime.h>
#include <hip/hip_bf16.h>

#define EPS 1e-5f

typedef __attribute__((ext_vector_type(2))) float f32x2;
typedef __attribute__((ext_vector_type(8))) float f32x8;

// Problem sizes (fixed by the reference)
#define Dsz 128
#define Hsz 64
#define Rsz 32
#define Tsz 8

// ws / LDS parameter block layout (float indices). ws layout == LDS layout so
// the main kernel does one linear copy.
#define OFF_BFRAG 0       // 8192 floats : stage-1 weights pre-swizzled into WMMA B-frag layout
#define OFF_FB    8192    // 64          : stage-1 bias
#define OFF_TW1T  8256    // 16384       : folded head weights, layout [t][j][r] (= tw1[t][r][j]*a1[j])
#define OFF_B2F   24640   // 256         : folded head bias  b'[t][r]
#define OFF_W2P   24896   // 256         : folded output weights w2'[t][r]
#define OFF_CC    25152   // 8           : per-head output constant
#define PARAM_F   25160   // total param floats (divisible by 4)

#define H_PAD     68                  // padded row stride for h tile (bank-conflict-free)
#define H_WAVE_F  (16 * H_PAD)        // 1088 floats per wave
#define SMEM_F    (PARAM_F + 8 * H_WAVE_F)   // 33864 floats = 135456 bytes

#define PERSISTENT_GRID 1024          // grid-stride blocks; params loaded once per block

// CDNA5 async direct-to-LDS copy path (bypasses VGPRs, tracked by ASYNCcnt).
#if defined(__gfx1250__) && \
    __has_builtin(__builtin_amdgcn_global_load_async_to_lds_b128) && \
    __has_builtin(__builtin_amdgcn_s_wait_asynccnt)
#define USE_ASYNC_LDS 1
// Builtin parameter types per hipcc diagnostic: pointer to 16-byte int vector;
// source in global (AS1), destination in LDS (AS3).
typedef int v4i_b128 __attribute__((vector_size(4 * sizeof(int))));
typedef __attribute__((address_space(1))) v4i_b128 gbl_v4i;
typedef __attribute__((address_space(3))) v4i_b128 lds_v4i;
#endif

// ---------------------------------------------------------------------------
// Prep kernel: fold BN into weights, pre-swizzle stage-1 weights into the
// exact VGPR lane layout of V_WMMA_F32_16X16X4_F32's B operand.
// ---------------------------------------------------------------------------
__global__ __launch_bounds__(256) void tarnet_prep(
    const float* __restrict__ fw, const float* __restrict__ fb,
    const float* __restrict__ f_gamma, const float* __restrict__ f_beta,
    const float* __restrict__ f_mean, const float* __restrict__ f_var,
    const float* __restrict__ tw1, const float* __restrict__ tb1,
    const float* __restrict__ t_gamma, const float* __restrict__ t_beta,
    const float* __restrict__ t_mean, const float* __restrict__ t_var,
    const float* __restrict__ tw2, const float* __restrict__ tb2,
    float* __restrict__ ws)
{
    __shared__ float a1[Hsz], c1[Hsz], tmp[Tsz * Rsz];
    const int tid = threadIdx.x;

    if (tid < Hsz) {
        float a = f_gamma[tid] * rsqrtf(f_var[tid] + EPS);
        a1[tid] = a;
        c1[tid] = f_beta[tid] - f_mean[tid] * a;
        ws[OFF_FB + tid] = fb[tid];
    }
    __syncthreads();

    // Stage-1 weight B-fragments. frag f = (k0/4)*4 + n0/16; 64 floats each.
    // Lane L<16 holds (B[k0][n0+L], B[k0+1][n0+L]); lanes 16-31 hold K+2/K+3.
    // B[k][n] = fw[n][k]  (h = x @ fw.T)
    for (int e = tid; e < 8192; e += 256) {
        int f = e >> 6, sl = e & 63, L = sl >> 1, half = sl & 1;
        int k0 = (f >> 2) << 2, n0 = (f & 3) << 4;
        int row, col;
        if (L < 16) { row = n0 + L;        col = k0 + half;     }
        else        { row = n0 + (L - 16); col = k0 + 2 + half; }
        ws[OFF_BFRAG + e] = fw[row * Dsz + col];
    }

    // Folded head weights, transposed to [t][j][r] for conflict-free lane-r reads.
    for (int e = tid; e < Tsz * Hsz * Rsz; e += 256) {
        int t = e >> 11, rem = e & 2047, j = rem >> 5, r = rem & 31;
        ws[OFF_TW1T + e] = tw1[t * (Rsz * Hsz) + r * Hsz + j] * a1[j];
    }

    {   // exactly 256 (t,r) pairs, one per thread
        const int e = tid;
        const int t = e >> 5;
        float sb = tb1[e];
        const float* wrow = tw1 + t * (Rsz * Hsz) + (e & 31) * Hsz;
        #pragma unroll
        for (int j = 0; j < Hsz; ++j) sb = __builtin_fmaf(wrow[j], c1[j], sb);
        ws[OFF_B2F + e] = sb;

        float a2 = t_gamma[e] * rsqrtf(t_var[e] + EPS);
        float c2 = t_beta[e] - t_mean[e] * a2;
        ws[OFF_W2P + e] = tw2[e] * a2;
        tmp[e] = tw2[e] * c2;
    }
    __syncthreads();

    if (tid < Tsz) {
        float sc = tb2[tid];
        #pragma unroll
        for (int r = 0; r < Rsz; ++r) sc += tmp[tid * Rsz + r];
        ws[OFF_CC + tid] = sc;
    }
}

// ---------------------------------------------------------------------------
// Main kernel: persistent blocks, 8 waves/block, 16 batch rows per wave per
// chunk. Stage 1: 128x v_wmma_f32_16x16x4_f32 per wave -> relu(h) tile in LDS.
// Stage 2: selected-head matvec per sample, wave32 shuffle reduction.
// Occupancy is LDS-bound at 2 blocks/WGP (4 waves/SIMD): allow the register
// allocator to use the full per-wave budget for deep load pipelining.
// ---------------------------------------------------------------------------
__global__ __launch_bounds__(256)
__attribute__((amdgpu_waves_per_eu(4)))
void tarnet_main(
    const float* __restrict__ x, const int* __restrict__ tsel,
    const float* __restrict__ ws, float* __restrict__ out, int nchunks)
{
    extern __shared__ float smem[];
    const int tid = threadIdx.x;

    // Load the parameter block into LDS once per (persistent) block.
#if defined(USE_ASYNC_LDS)
    {
        gbl_v4i* src = (gbl_v4i*)ws;
        lds_v4i* dst = (lds_v4i*)smem;
        for (int i = tid; i < PARAM_F / 4; i += 256)
            __builtin_amdgcn_global_load_async_to_lds_b128(src + i, dst + i, 0, 0);
        __builtin_amdgcn_s_wait_asynccnt(0);
    }
#else
    {
        const float4* src = (const float4*)ws;
        float4* dst = (float4*)smem;
        for (int i = tid; i < PARAM_F / 4; i += 256) dst[i] = src[i];
    }
#endif
    __syncthreads();

    const int wave  = tid >> 5;
    const int lane  = tid & 31;
    const int lrow  = lane & 15;   // M within half / N within tile
    const int lhalf = lane >> 4;   // selects K pair (A) / K+2 (B) / M+8 (D)
    const f32x2* bfrag = (const f32x2*)(smem + OFF_BFRAG);
    float* hb = smem + PARAM_F + wave * H_WAVE_F;

    for (int chunk = blockIdx.x; chunk < nchunks; chunk += gridDim.x) {
        const int m0 = chunk * 128 + wave * 16;

        // A-operand source: lane L<16 -> row m0+L, cols (k0,k0+1);
        //                   lane L>=16 -> row m0+L-16, cols (k0+2,k0+3).
        const f32x2* xv = (const f32x2*)(x + (size_t)(m0 + lrow) * Dsz + 2 * lhalf);

        // Preload the whole K=128 strip of A-fragments in one load clause so
        // waits amortize over the 128-WMMA stream (instead of wait-to-zero
        // before every quad).
        f32x2 areg[Dsz / 4];
        #pragma unroll
        for (int kq = 0; kq < Dsz / 4; ++kq) areg[kq] = xv[2 * kq];

        f32x8 acc[4];
        {
            f32x8 zero = {0.f, 0.f, 0.f, 0.f, 0.f, 0.f, 0.f, 0.f};
            #pragma unroll
            for (int nt = 0; nt < 4; ++nt) acc[nt] = zero;
        }

        #pragma unroll
        for (int kq = 0; kq < Dsz / 4; ++kq) {
            #pragma unroll
            for (int nt = 0; nt < 4; ++nt) {
                f32x2 b = bfrag[(size_t)(kq * 4 + nt) * 32 + lane];
                acc[nt] = __builtin_amdgcn_wmma_f32_16x16x4_f32(
                    false, areg[kq], false, b, (short)0, acc[nt], false, false);
            }
        }

        // Bias + ReLU, store h tile to wave-private padded LDS region.
        #pragma unroll
        for (int nt = 0; nt < 4; ++nt) {
            float fbv = smem[OFF_FB + nt * 16 + lrow];
            #pragma unroll
            for (int r = 0; r < 8; ++r) {
                float v = acc[nt][r] + fbv;       // D: VGPR r -> M=r(+8), N=lane&15
                v = v > 0.f ? v : 0.f;
                hb[(r + 8 * lhalf) * H_PAD + nt * 16 + lrow] = v;
            }
        }

        // Stage 2: only the selected head per sample. lane == r.
        #pragma unroll 1
        for (int s = 0; s < 16; ++s) {
            const int th = tsel[m0 + s];
            const float* w  = smem + OFF_TW1T + th * (Hsz * Rsz) + lane;  // [j][r]
            const float* hv = hb + s * H_PAD;
            float z = smem[OFF_B2F + th * Rsz + lane];
            #pragma unroll
            for (int j = 0; j < Hsz; ++j)
                z = __builtin_fmaf(hv[j], w[j * Rsz], z);
            z = (z > 0.f ? z : 0.f) * smem[OFF_W2P + th * Rsz + lane];
            #pragma unroll
            for (int off = 16; off > 0; off >>= 1)
                z += __shfl_xor(z, off, 32);
            if (lane == 0) out[m0 + s] = z + smem[OFF_CC + th];
        }
        // h region is wave-private: no block barrier needed between chunks.
    }
}

// ---------------------------------------------------------------------------
extern "C" void kernel_launch(void* const* d_in, const int* in_sizes, int n_in,
                              void* d_out, int out_size, void* d_ws, size_t ws_size,
                              hipStream_t stream)
{
    const float* x       = (const float*)d_in[0];
    const int*   t       = (const int*)  d_in[1];
    const float* fw      = (const float*)d_in[2];
    const float* fb      = (const float*)d_in[3];
    const float* f_gamma = (const float*)d_in[4];
    const float* f_beta  = (const float*)d_in[5];
    const float* f_mean  = (const float*)d_in[6];
    const float* f_var   = (const float*)d_in[7];
    const float* tw1     = (const float*)d_in[8];
    const float* tb1     = (const float*)d_in[9];
    const float* t_gamma = (const float*)d_in[10];
    const float* t_beta  = (const float*)d_in[11];
    const float* t_mean  = (const float*)d_in[12];
    const float* t_var   = (const float*)d_in[13];
    const float* tw2     = (const float*)d_in[14];
    const float* tb2     = (const float*)d_in[15];
    float* ws  = (float*)d_ws;     // needs PARAM_F*4 = 100640 bytes
    float* out = (float*)d_out;

    tarnet_prep<<<1, 256, 0, stream>>>(fw, fb, f_gamma, f_beta, f_mean, f_var,
                                       tw1, tb1, t_gamma, t_beta, t_mean, t_var,
                                       tw2, tb2, ws);

    const int B = in_sizes[0] / Dsz;       // 262144
    const int nchunks = B / 128;           // 2048 chunks of 128 rows
    const int grid = (nchunks < PERSISTENT_GRID) ? nchunks : PERSISTENT_GRID;
    const size_t smem_bytes = (size_t)SMEM_F * sizeof(float);  // 135456 B (<320KB/WGP)
    (void)hipFuncSetAttribute(reinterpret_cast<const void*>(tarnet_main),
                              hipFuncAttributeMaxDynamicSharedMemorySize,
                              (int)smem_bytes);
    tarnet_main<<<grid, 256, smem_bytes, stream>>>(x, t, ws, out, nchunks);
}